// GMMLoss2_15315853377592
// MI455X (gfx1250) — compile-verified
//
#include <hip/hip_runtime.h>

#define BATCH 8192
#define KDIM  64
#define NCLS  8
#define NBLK  32                  // stats blocks
#define ROWS_PER_BLK (BATCH / NBLK)

typedef __attribute__((ext_vector_type(2))) float v2f;
typedef __attribute__((ext_vector_type(8))) float v8f;

// ---------------------------------------------------------------------------
// Kernel 1: deterministic per-block partial stats (class sums, counts, first pos)
// blockDim = 64 (thread t owns feature column t), grid = NBLK
// ---------------------------------------------------------------------------
__global__ __launch_bounds__(64) void stats_kernel(const float* __restrict__ mu,
                                                   const int* __restrict__ lab,
                                                   float* __restrict__ psums,  // [NBLK][NCLS][KDIM]
                                                   int* __restrict__ pcnt,     // [NBLK][NCLS]
                                                   int* __restrict__ pfpos) {  // [NBLK][NCLS]
    const int t   = threadIdx.x;       // feature index 0..63
    const int blk = blockIdx.x;
    const int b0  = blk * ROWS_PER_BLK;

    float acc[NCLS];
#pragma unroll
    for (int c = 0; c < NCLS; ++c) acc[c] = 0.0f;
    int cnt[NCLS], fp[NCLS];
#pragma unroll
    for (int c = 0; c < NCLS; ++c) { cnt[c] = 0; fp[c] = BATCH; }

    for (int r = 0; r < ROWS_PER_BLK; ++r) {
        const int b = b0 + r;
        const int c = lab[b];
        const float v = mu[b * KDIM + t];
#pragma unroll
        for (int cc = 0; cc < NCLS; ++cc) acc[cc] += (cc == c) ? v : 0.0f;
        if (t == 0) {
            cnt[c]++;
            if (fp[c] == BATCH) fp[c] = b;
        }
    }
#pragma unroll
    for (int c = 0; c < NCLS; ++c)
        psums[(blk * NCLS + c) * KDIM + t] = acc[c];
    if (t == 0) {
#pragma unroll
        for (int c = 0; c < NCLS; ++c) {
            pcnt[blk * NCLS + c]  = cnt[c];
            pfpos[blk * NCLS + c] = fp[c];
        }
    }
}

// ---------------------------------------------------------------------------
// Kernel 2: class-masked Gram matrices G_c = sum_{b: lab[b]==c} mu_b mu_b^T
// via V_WMMA_F32_16X16X4_F32.  grid = NCLS*16 (class, 16x16 output tile),
// blockDim = 128 (4 waves partition the batch, LDS reduce, direct store).
// ---------------------------------------------------------------------------
__global__ __launch_bounds__(128) void gram_kernel(const float* __restrict__ mu,
                                                   const int* __restrict__ lab,
                                                   float* __restrict__ G) {   // [NCLS][64][64]
    const int cls  = blockIdx.x >> 4;
    const int tile = blockIdx.x & 15;
    const int ti   = (tile >> 2) * 16;     // output row block
    const int tj   = (tile & 3) * 16;      // output col block

    const int tid  = threadIdx.x;
    const int wave = tid >> 5;
    const int lane = tid & 31;
    const int half = lane >> 4;            // 0: K={0,1}, 1: K={2,3}
    const int idx  = lane & 15;            // M (A) / N (B) index

    v8f acc0 = {};
    v8f acc1 = {};

    // each wave owns two 4-row chunks per iteration; 4 waves -> stride 32 rows
    for (int b0 = wave * 8; b0 < BATCH; b0 += 32) {
        {
            const int r  = b0 + 2 * half;
            const float m0 = (lab[r]     == cls) ? 1.0f : 0.0f;
            const float m1 = (lab[r + 1] == cls) ? 1.0f : 0.0f;
            v2f a, b;
            a.x = mu[r * KDIM + ti + idx] * m0;           // A[m,k]: masked mu^T tile
            a.y = mu[(r + 1) * KDIM + ti + idx] * m1;
            b.x = mu[r * KDIM + tj + idx];                // B[k,n]: mu tile
            b.y = mu[(r + 1) * KDIM + tj + idx];
            acc0 = __builtin_amdgcn_wmma_f32_16x16x4_f32(
                false, a, false, b, (short)0, acc0, false, false);
        }
        {
            const int r  = b0 + 4 + 2 * half;
            const float m0 = (lab[r]     == cls) ? 1.0f : 0.0f;
            const float m1 = (lab[r + 1] == cls) ? 1.0f : 0.0f;
            v2f a, b;
            a.x = mu[r * KDIM + ti + idx] * m0;
            a.y = mu[(r + 1) * KDIM + ti + idx] * m1;
            b.x = mu[r * KDIM + tj + idx];
            b.y = mu[(r + 1) * KDIM + tj + idx];
            acc1 = __builtin_amdgcn_wmma_f32_16x16x4_f32(
                false, a, false, b, (short)0, acc1, false, false);
        }
    }

    __shared__ float red[4][32][8];
#pragma unroll
    for (int r = 0; r < 8; ++r) red[wave][lane][r] = acc0[r] + acc1[r];
    __syncthreads();

    if (wave == 0) {
#pragma unroll
        for (int r = 0; r < 8; ++r) {
            const float s = red[0][lane][r] + red[1][lane][r] +
                            red[2][lane][r] + red[3][lane][r];
            // C/D layout: VGPR r, lanes 0-15 -> M=r, lanes 16-31 -> M=r+8; N=idx
            const int M = r + 8 * half;
            G[(cls * KDIM + (ti + M)) * KDIM + (tj + idx)] = s;
        }
    }
}

// ---------------------------------------------------------------------------
// Small block-parallel linear algebra helpers (blockDim == 64)
// ---------------------------------------------------------------------------
__device__ void chol64(float (*M)[KDIM + 1], int t) {
    for (int k = 0; k < KDIM; ++k) {
        if (t == k) M[k][k] = sqrtf(M[k][k]);
        __syncthreads();
        const float dk = M[k][k];
        if (t > k) M[t][k] /= dk;
        __syncthreads();
        if (t > k) {
            const float lik = M[t][k];
            for (int j = k + 1; j <= t; ++j) M[t][j] -= lik * M[j][k];
        }
        __syncthreads();
    }
}

__device__ float block_sum64(float v, int t, float* red) {
    red[t] = v;
    __syncthreads();
    float s = 0.0f;
    for (int i = 0; i < KDIM; ++i) s += red[i];
    __syncthreads();
    return s;
}

// ---------------------------------------------------------------------------
// Kernel 3: finalize — build Sigmas, Choleskys, KL terms, weighting, scalar out
// 1 block x 64 threads (thread t owns matrix column t)
// ---------------------------------------------------------------------------
__global__ __launch_bounds__(64) void finalize_kernel(const float* __restrict__ G,
                                                      const float* __restrict__ psums,
                                                      const int* __restrict__ pcnt,
                                                      const int* __restrict__ pfpos,
                                                      const float* __restrict__ cov,
                                                      float* __restrict__ out) {
    __shared__ float S[KDIM][KDIM + 1];   // Sigma_total -> its Cholesky L
    __shared__ float T[KDIM][KDIM + 1];   // Sigma_F[c]  -> Cholesky R -> L^-1 R
    __shared__ float muF[NCLS][KDIM];
    __shared__ float muT[KDIM];
    __shared__ float red[KDIM];
    __shared__ float y[KDIM];
    __shared__ float klv[NCLS];
    __shared__ float cnts[NCLS];
    __shared__ int   fposs[NCLS];

    const int t = threadIdx.x;  // 0..63

    if (t == 0) {
        for (int c = 0; c < NCLS; ++c) {
            int cc = 0, fp = BATCH;
            for (int blk = 0; blk < NBLK; ++blk) {
                cc += pcnt[blk * NCLS + c];
                const int f = pfpos[blk * NCLS + c];
                fp = (f < fp) ? f : fp;
            }
            cnts[c]  = (float)cc;
            fposs[c] = fp;
        }
    }
    __syncthreads();

    // group means + total mean (fixed order -> deterministic)
    float tot = 0.0f;
    for (int c = 0; c < NCLS; ++c) {
        float s = 0.0f;
        for (int blk = 0; blk < NBLK; ++blk)
            s += psums[(blk * NCLS + c) * KDIM + t];
        muF[c][t] = s / fmaxf(cnts[c], 1.0f);
        tot += s;
    }
    muT[t] = tot / (float)BATCH;
    __syncthreads();

    // Sigma_total = cov + G_total/B - muT muT^T   (column t)
    for (int i = 0; i < KDIM; ++i) {
        float g = 0.0f;
        for (int c = 0; c < NCLS; ++c) g += G[(c * KDIM + i) * KDIM + t];
        S[i][t] = cov[i * KDIM + t] + g / (float)BATCH - muT[i] * muT[t];
    }
    __syncthreads();

    chol64(S, t);
    const float ld2 = 2.0f * block_sum64(logf(S[t][t]), t, red);

    for (int c = 0; c < NCLS; ++c) {
        const float cnt  = cnts[c];
        const float safe = fmaxf(cnt, 1.0f);
        // Sigma_F[c] = cov + (G_c - cnt * muF muF^T)/safe
        for (int i = 0; i < KDIM; ++i)
            T[i][t] = cov[i * KDIM + t] +
                      (G[(c * KDIM + i) * KDIM + t] - cnt * muF[c][i] * muF[c][t]) / safe;
        __syncthreads();

        chol64(T, t);  // lower triangle of T = R,  Sigma_F = R R^T
        const float ld1 = 2.0f * block_sum64(logf(T[t][t]), t, red);

        // tr(inv(S2) S1) = || L^-1 R ||_F^2 ; W = L^-1 R is lower triangular,
        // so column t needs only rows i >= t (in-place forward substitution).
        float trc = 0.0f;
        for (int i = t; i < KDIM; ++i) {
            float v = T[i][t];
            for (int m = t; m < i; ++m) v -= S[i][m] * T[m][t];
            v /= S[i][i];
            T[i][t] = v;
            trc += v * v;
        }
        const float tr = block_sum64(trc, t, red);

        if (t == 0) {
            // quad = d^T inv(S2) d = || L^-1 d ||^2
            float q = 0.0f;
            for (int i = 0; i < KDIM; ++i) {
                float v = muT[i] - muF[c][i];
                for (int m = 0; m < i; ++m) v -= S[i][m] * y[m];
                v /= S[i][i];
                y[i] = v;
                q += v * v;
            }
            klv[c] = 0.5f * (tr + q - (float)KDIM + ld2 - ld1);
        }
        __syncthreads();
    }

    if (t == 0) {
        // last_key = argmax over present classes of first position (first max on ties)
        int last = 0, best = -1;
        for (int c = 0; c < NCLS; ++c) {
            const int v = (fposs[c] < BATCH) ? fposs[c] : -1;
            if (v > best) { best = v; last = c; }
        }
        float s = 0.0f;
        for (int c = 0; c < NCLS; ++c) {
            const bool present = fposs[c] < BATCH;
            const float w = (present && c != last) ? cnts[c] : 0.0f;
            s += klv[c] * w;
        }
        out[0] = s / (float)BATCH;   // sum(kl*w*B)/B^2
    }
}

// ---------------------------------------------------------------------------
// Host launcher
// ---------------------------------------------------------------------------
extern "C" void kernel_launch(void* const* d_in, const int* in_sizes, int n_in,
                              void* d_out, int out_size, void* d_ws, size_t ws_size,
                              hipStream_t stream) {
    const float* mu  = (const float*)d_in[0];
    const int*   lab = (const int*)d_in[1];
    const float* cov = (const float*)d_in[2];
    float* out = (float*)d_out;

    // workspace layout (all 4-byte aligned)
    float* G     = (float*)d_ws;                       // NCLS*64*64
    float* psums = G + NCLS * KDIM * KDIM;             // NBLK*NCLS*KDIM
    int*   pcnt  = (int*)(psums + NBLK * NCLS * KDIM); // NBLK*NCLS
    int*   pfpos = pcnt + NBLK * NCLS;                 // NBLK*NCLS

    stats_kernel<<<NBLK, 64, 0, stream>>>(mu, lab, psums, pcnt, pfpos);
    gram_kernel<<<NCLS * 16, 128, 0, stream>>>(mu, lab, G);
    finalize_kernel<<<1, 64, 0, stream>>>(G, psums, pcnt, pfpos, cov, out);
}